// STEWeights_15771119911212
// MI455X (gfx1250) — compile-verified
//
#include <hip/hip_runtime.h>
#include <hip/hip_bf16.h>

// out[b,o] = sum_k x[b,k] * tern(W[o,k]),  tern(w) = sign(w)*(|w|>=0.33)
// M=8192, K=4096, N=16384.
// Pass 1: convert x->bf16, W->ternary bf16 into workspace (bandwidth bound).
// Pass 2: bf16 WMMA GEMM, f32 accumulation; tiles staged global->LDS with
//         CDNA5 async copies (global_load_async_to_lds_b128, ASYNCcnt).
//         WMMA as inline asm with C/D tied ("+v"); k-loop kept at unroll 1 so
//         accumulator chains coalesce into fixed tuples (no v_mov storm).
// Fused single-pass fallback if workspace is too small.

typedef __attribute__((ext_vector_type(16))) __bf16 v16bf;
typedef __attribute__((ext_vector_type(8)))  __bf16 v8bf;
typedef __attribute__((ext_vector_type(4)))  __bf16 v4bf;
typedef __attribute__((ext_vector_type(8)))  float  v8f;

#define M_DIM 8192
#define K_DIM 4096
#define N_DIM 16384

#define TM 128
#define TN 128
#define TK 32
#define LDK (TK + 8)   // 80B rows: 16B-aligned, spreads LDS banks

__device__ __forceinline__ __bf16 tern(float v) {
    float h = (fabsf(v) >= 0.33f) ? copysignf(1.0f, v) : 0.0f;
    return (__bf16)h;
}

// D(=C) += A x B, C/D tied in one tuple. No WMMA->WMMA hazard: D only feeds
// the next C (hazard table covers only D->A/B), never A/B.
__device__ __forceinline__ void wmma_bf16(v8f& acc, v16bf a, v16bf b) {
    asm("v_wmma_f32_16x16x32_bf16 %0, %1, %2, %0"
        : "+v"(acc)
        : "v"(a), "v"(b));
}

// WMMA_*BF16 -> VALU hazard: 4 co-exec NOPs before VALU touches D.
__device__ __forceinline__ void wmma_valu_fence() {
    asm volatile("s_nop 4" ::: "memory");
}

// ---------------------------------------------------------------- pass 1 ----
__global__ __launch_bounds__(256)
void cvt_x_bf16(const float* __restrict__ in, __bf16* __restrict__ out) {
    size_t i = ((size_t)blockIdx.x * 256 + threadIdx.x) * 8;
    float4 a = *(const float4*)(in + i);
    float4 b = *(const float4*)(in + i + 4);
    v8bf o = { (__bf16)a.x, (__bf16)a.y, (__bf16)a.z, (__bf16)a.w,
               (__bf16)b.x, (__bf16)b.y, (__bf16)b.z, (__bf16)b.w };
    *(v8bf*)(out + i) = o;
}

__global__ __launch_bounds__(256)
void tern_w_bf16(const float* __restrict__ in, __bf16* __restrict__ out) {
    size_t i = ((size_t)blockIdx.x * 256 + threadIdx.x) * 8;
    float4 a = *(const float4*)(in + i);
    float4 b = *(const float4*)(in + i + 4);
    v8bf o = { tern(a.x), tern(a.y), tern(a.z), tern(a.w),
               tern(b.x), tern(b.y), tern(b.z), tern(b.w) };
    *(v8bf*)(out + i) = o;
}

// ---------------------------------------------------------------- pass 2 ----
__device__ __forceinline__ void async_copy16(uint32_t lds_off, const __bf16* g) {
    // 16B global -> LDS, tracked by ASYNCcnt (CDNA5 async-copy path)
    asm volatile("global_load_async_to_lds_b128 %0, %1, off"
                 :: "v"(lds_off), "v"(g) : "memory");
}
__device__ __forceinline__ void wait_async0() {
    asm volatile("s_wait_asynccnt 0x0" ::: "memory");
}

__global__ __launch_bounds__(256, 2)
void ste_gemm_bf16(const __bf16* __restrict__ xa,   // [M,K] bf16, k-contig
                   const __bf16* __restrict__ wb,   // [N,K] bf16 ternary
                   float* __restrict__ out)
{
    __shared__ __bf16 As[2][TM][LDK];   // 10 KB per buffer
    __shared__ __bf16 Bs[2][TN][LDK];

    const int tid  = threadIdx.x;
    const int lane = tid & 31;
    const int lrow = lane & 15;   // M (A/C) or N (B) index within fragment
    const int lhi  = lane >> 4;   // half-wave selector
    const int wave = tid >> 5;
    const int wm   = wave & 3;    // 4 waves along M, 32 rows each
    const int wn   = wave >> 2;   // 2 waves along N, 64 cols each

    const int m0 = blockIdx.y * TM;
    const int n0 = blockIdx.x * TN;

    // Staging: tile 128 rows x 32 bf16 (64B/row); 4 threads/row x 16B,
    // 64 rows per pass, 2 passes -> 4 async b128 per thread per tile.
    const int sr = tid >> 2;            // 0..63
    const int sc = (tid & 3) << 3;      // 0,8,16,24 (bf16 elems)

    const __bf16* pA0 = xa + (size_t)(m0 + sr)      * K_DIM + sc;
    const __bf16* pA1 = xa + (size_t)(m0 + sr + 64) * K_DIM + sc;
    const __bf16* pB0 = wb + (size_t)(n0 + sr)      * K_DIM + sc;
    const __bf16* pB1 = wb + (size_t)(n0 + sr + 64) * K_DIM + sc;

    // LDS byte offsets of this thread's staging slots (low 32 bits of flat
    // address == LDS offset). All 16B aligned (row stride 80B = 5*16).
    const uint32_t aoff0 = (uint32_t)(uintptr_t)&As[0][sr][sc];
    const uint32_t boff0 = (uint32_t)(uintptr_t)&Bs[0][sr][sc];
    const uint32_t bufStride = (uint32_t)(TM * LDK * sizeof(__bf16));   // 10240
    const uint32_t rowBlk    = (uint32_t)(64 * LDK * sizeof(__bf16));   // 5120

    auto issueAsync = [&](int buf) {
        const uint32_t a0 = aoff0 + (uint32_t)buf * bufStride;
        const uint32_t b0 = boff0 + (uint32_t)buf * bufStride;
        async_copy16(a0,          pA0);
        async_copy16(a0 + rowBlk, pA1);
        async_copy16(b0,          pB0);
        async_copy16(b0 + rowBlk, pB1);
    };

    v8f acc[2][4];
    #pragma unroll
    for (int mf = 0; mf < 2; ++mf)
        #pragma unroll
        for (int nf = 0; nf < 4; ++nf)
            acc[mf][nf] = {};

    auto compute = [&](int buf) {
        v16bf afrag[2], bfrag[4];
        // A fragment (16x32): lane<16  -> M=lane,    K = 0..7  & 16..23
        //                     lane>=16 -> M=lane-16, K = 8..15 & 24..31
        #pragma unroll
        for (int mf = 0; mf < 2; ++mf) {
            const __bf16* p = &As[buf][wm * 32 + mf * 16 + lrow][lhi * 8];
            uint4* q = (uint4*)&afrag[mf];
            q[0] = *(const uint4*)(p);
            q[1] = *(const uint4*)(p + 16);
        }
        // B fragment (32x16): lane<16  -> N=lane,    K = 0..15 contiguous
        //                     lane>=16 -> N=lane-16, K = 16..31 contiguous
        #pragma unroll
        for (int nf = 0; nf < 4; ++nf) {
            const __bf16* p = &Bs[buf][wn * 64 + nf * 16 + lrow][lhi * 16];
            uint4* q = (uint4*)&bfrag[nf];
            q[0] = *(const uint4*)(p);
            q[1] = *(const uint4*)(p + 8);
        }
        #pragma unroll
        for (int mf = 0; mf < 2; ++mf)
            #pragma unroll
            for (int nf = 0; nf < 4; ++nf)
                wmma_bf16(acc[mf][nf], afrag[mf], bfrag[nf]);
    };

    // Prologue: stage tile 0
    issueAsync(0);
    wait_async0();
    __syncthreads();

    const int NKT = K_DIM / TK;   // 128
    #pragma unroll 1
    for (int t = 0; t < NKT; ++t) {
        const int buf = t & 1;
        if (t + 1 < NKT) {
            pA0 += TK; pA1 += TK; pB0 += TK; pB1 += TK;
            issueAsync(buf ^ 1);   // safe: buf^1 reads finished before last barrier
        }
        compute(buf);
        if (t + 1 < NKT) {
            wait_async0();         // this wave's async writes landed in LDS
            __syncthreads();       // all waves' writes visible
        }
    }

    wmma_valu_fence();

    // C layout: VGPR j -> M=j (lanes 0-15) / M=j+8 (lanes 16-31); N=lane%16
    #pragma unroll
    for (int mf = 0; mf < 2; ++mf)
        #pragma unroll
        for (int nf = 0; nf < 4; ++nf) {
            const int col = n0 + wn * 64 + nf * 16 + lrow;
            #pragma unroll
            for (int j = 0; j < 8; ++j) {
                const int row = m0 + wm * 32 + mf * 16 + lhi * 8 + j;
                out[(size_t)row * N_DIM + col] = acc[mf][nf][j];
            }
        }
}

// ---------------------------------------------- fused fallback (small ws) ----
__global__ __launch_bounds__(256, 2)
void ste_ternary_gemm_fused(const float* __restrict__ x,
                            const float* __restrict__ w,
                            float* __restrict__ out)
{
    __shared__ __bf16 As[2][TM][LDK];
    __shared__ __bf16 Bs[2][TN][LDK];

    const int tid  = threadIdx.x;
    const int lane = tid & 31;
    const int lrow = lane & 15;
    const int lhi  = lane >> 4;
    const int wave = tid >> 5;
    const int wm   = wave & 3;
    const int wn   = wave >> 2;

    const int m0 = blockIdx.y * TM;
    const int n0 = blockIdx.x * TN;

    const int lr = tid >> 3;
    const int lc = (tid & 7) << 2;

    float4 avr[4], bvr[4];

    auto loadTiles = [&](int kt) {
        const int k0 = kt * TK;
        #pragma unroll
        for (int i = 0; i < 4; ++i) {
            const int r = lr + 32 * i;
            avr[i] = *(const float4*)(x + (size_t)(m0 + r) * K_DIM + k0 + lc);
            bvr[i] = *(const float4*)(w + (size_t)(n0 + r) * K_DIM + k0 + lc);
        }
    };

    auto stageTiles = [&](int buf) {
        #pragma unroll
        for (int i = 0; i < 4; ++i) {
            const int r = lr + 32 * i;
            v4bf a = { (__bf16)avr[i].x, (__bf16)avr[i].y,
                       (__bf16)avr[i].z, (__bf16)avr[i].w };
            *(v4bf*)&As[buf][r][lc] = a;
            v4bf b = { tern(bvr[i].x), tern(bvr[i].y),
                       tern(bvr[i].z), tern(bvr[i].w) };
            *(v4bf*)&Bs[buf][r][lc] = b;
        }
    };

    v8f acc[2][4];
    #pragma unroll
    for (int mf = 0; mf < 2; ++mf)
        #pragma unroll
        for (int nf = 0; nf < 4; ++nf)
            acc[mf][nf] = {};

    auto compute = [&](int buf) {
        v16bf afrag[2], bfrag[4];
        #pragma unroll
        for (int mf = 0; mf < 2; ++mf) {
            const __bf16* p = &As[buf][wm * 32 + mf * 16 + lrow][lhi * 8];
            uint4* q = (uint4*)&afrag[mf];
            q[0] = *(const uint4*)(p);
            q[1] = *(const uint4*)(p + 16);
        }
        #pragma unroll
        for (int nf = 0; nf < 4; ++nf) {
            const __bf16* p = &Bs[buf][wn * 64 + nf * 16 + lrow][lhi * 16];
            uint4* q = (uint4*)&bfrag[nf];
            q[0] = *(const uint4*)(p);
            q[1] = *(const uint4*)(p + 8);
        }
        #pragma unroll
        for (int mf = 0; mf < 2; ++mf)
            #pragma unroll
            for (int nf = 0; nf < 4; ++nf)
                wmma_bf16(acc[mf][nf], afrag[mf], bfrag[nf]);
    };

    loadTiles(0);
    stageTiles(0);
    __syncthreads();

    const int NKT = K_DIM / TK;
    #pragma unroll 1
    for (int t = 0; t < NKT; ++t) {
        const int buf = t & 1;
        if (t + 1 < NKT) loadTiles(t + 1);
        compute(buf);
        if (t + 1 < NKT) {
            __syncthreads();
            stageTiles(buf ^ 1);
            __syncthreads();
        }
    }

    wmma_valu_fence();

    #pragma unroll
    for (int mf = 0; mf < 2; ++mf)
        #pragma unroll
        for (int nf = 0; nf < 4; ++nf) {
            const int col = n0 + wn * 64 + nf * 16 + lrow;
            #pragma unroll
            for (int j = 0; j < 8; ++j) {
                const int row = m0 + wm * 32 + mf * 16 + lhi * 8 + j;
                out[(size_t)row * N_DIM + col] = acc[mf][nf][j];
            }
        }
}

// --------------------------------------------------------------- launch ----
extern "C" void kernel_launch(void* const* d_in, const int* in_sizes, int n_in,
                              void* d_out, int out_size, void* d_ws, size_t ws_size,
                              hipStream_t stream) {
    (void)in_sizes; (void)n_in; (void)out_size;
    const float* x = (const float*)d_in[0];
    const float* w = (const float*)d_in[1];
    // d_in[2] (mask) does not affect the forward value: forward uses hard weights.
    float* out = (float*)d_out;

    const size_t nx = (size_t)M_DIM * K_DIM;          // 33.5M
    const size_t nw = (size_t)N_DIM * K_DIM;          // 67.1M
    const size_t need = (nx + nw) * sizeof(__bf16);   // ~192 MB

    dim3 grid(N_DIM / TN, M_DIM / TM);                // (128, 64)

    if (ws_size >= need) {
        __bf16* xb = (__bf16*)d_ws;
        __bf16* wb = xb + nx;
        cvt_x_bf16 <<<dim3(nx / (8 * 256)), dim3(256), 0, stream>>>(x, xb);
        tern_w_bf16<<<dim3(nw / (8 * 256)), dim3(256), 0, stream>>>(w, wb);
        ste_gemm_bf16<<<grid, dim3(256), 0, stream>>>(xb, wb, out);
    } else {
        ste_ternary_gemm_fused<<<grid, dim3(256), 0, stream>>>(x, w, out);
    }
}